// PostProcessor_77249281786350
// MI455X (gfx1250) — compile-verified
//
#include <hip/hip_runtime.h>
#include <stdint.h>

// ---------------- problem constants (match reference) ----------------
#define BATCH 2
#define NPROP 1000
#define NCLS  81
#define NC1   80                 // foreground classes
#define MSORT 1024               // sort size (pow2 >= NPROP)
#define TPB   512                // threads per NMS workgroup (16 waves of 32)

#define IMG_W 1333.0f
#define IMG_H 800.0f
#define SCORE_THRESH 0.05f
#define NMS_THRESH   0.5f
#define BBOX_CLIP    4.135166556742356f   // ln(1000/16)

typedef unsigned int u32x4_t __attribute__((ext_vector_type(4)));
typedef unsigned int u32x8_t __attribute__((ext_vector_type(8)));

// Generic (flat) pointer to LDS: low 32 bits of the flat address are the LDS
// byte offset (CDNA5 ISA).
__device__ __forceinline__ unsigned lds_off(const void* p) {
  return (unsigned)(unsigned long long)p;
}

// Per-lane async gather: copies 16 bytes from the global address in vaddr to the LDS address in vdst, ASYNCcnt.
__device__ __forceinline__ void async_gather_b128(const void* gaddr, void* laddr) {
  unsigned l = lds_off(laddr);
  asm volatile("global_load_async_to_lds_b128 %0, %1, off"
               :: "v"(l), "v"(gaddr) : "memory");
}

__device__ __forceinline__ void wait_async0() {
#if __has_builtin(__builtin_amdgcn_s_wait_asynccnt)
  __builtin_amdgcn_s_wait_asynccnt(0);
#else
  asm volatile("s_wait_asynccnt 0" ::: "memory");
#endif
}

__device__ __forceinline__ void wait_tensor0() {
#if __has_builtin(__builtin_amdgcn_s_wait_tensorcnt)
  __builtin_amdgcn_s_wait_tensorcnt(0);
#else
  asm volatile("s_wait_tensorcnt 0" ::: "memory");
#endif
}

// ======================================================================
// Kernel A: softmax over 81 classes + box decode + clip, one wave per row.
// Writes class-major scratch so the NMS kernel streams contiguously.
//   scores_ws: [B*80, 1000]      boxes_ws: [B*80, 1000, 4]
// ======================================================================
__global__ __launch_bounds__(128) void decode_softmax_kernel(
    const float* __restrict__ logits,     // [2000, 81]
    const float* __restrict__ regr,       // [2000, 324]
    const float* __restrict__ props,      // [2000, 4]
    float* __restrict__ scores_ws,
    float* __restrict__ boxes_ws)
{
  const int lane = threadIdx.x & 31;
  const int row  = blockIdx.x * 4 + (threadIdx.x >> 5);
  if (row >= BATCH * NPROP) return;
  const int b = row / NPROP;
  const int n = row - b * NPROP;

  // ---- softmax over 81 logits, 3 per lane, wave32 butterfly reductions
  const float* lrow = logits + (size_t)row * NCLS;
  float l0 = lrow[lane];
  float l1 = lrow[lane + 32];
  float l2 = (lane < NCLS - 64) ? lrow[lane + 64] : -1.0e30f;

  float m = fmaxf(l0, fmaxf(l1, l2));
  #pragma unroll
  for (int off = 16; off > 0; off >>= 1) m = fmaxf(m, __shfl_xor(m, off, 32));

  float e0 = __expf(l0 - m);
  float e1 = __expf(l1 - m);
  float e2 = (lane < NCLS - 64) ? __expf(l2 - m) : 0.0f;
  float s = e0 + e1 + e2;
  #pragma unroll
  for (int off = 16; off > 0; off >>= 1) s += __shfl_xor(s, off, 32);
  const float inv = 1.0f / s;

  // ---- proposal geometry (broadcast across the wave)
  const float4 pb = ((const float4*)props)[row];
  const float w  = pb.z - pb.x + 1.0f;
  const float h  = pb.w - pb.y + 1.0f;
  const float cx = pb.x + 0.5f * w;
  const float cy = pb.y + 0.5f * h;

  const float4* rrow = (const float4*)(regr + (size_t)row * (4 * NCLS));
  float ev[3] = {e0, e1, e2};

  #pragma unroll
  for (int k = 0; k < 3; ++k) {
    int c = lane + 32 * k;
    if (c >= 1 && c < NCLS) {                 // skip background class 0
      float prob = ev[k] * inv;
      float4 rg = rrow[c];
      float dx = rg.x * 0.1f;                 // /WX
      float dy = rg.y * 0.1f;                 // /WY
      float dw = fminf(rg.z * 0.2f, BBOX_CLIP);
      float dh = fminf(rg.w * 0.2f, BBOX_CLIP);
      float pcx = fmaf(dx, w, cx);
      float pcy = fmaf(dy, h, cy);
      float pw = __expf(dw) * w;
      float ph = __expf(dh) * h;
      float x1 = pcx - 0.5f * pw;
      float y1 = pcy - 0.5f * ph;
      float x2 = pcx + 0.5f * pw - 1.0f;
      float y2 = pcy + 0.5f * ph - 1.0f;
      x1 = fminf(fmaxf(x1, 0.0f), IMG_W - 1.0f);
      y1 = fminf(fmaxf(y1, 0.0f), IMG_H - 1.0f);
      x2 = fminf(fmaxf(x2, 0.0f), IMG_W - 1.0f);
      y2 = fminf(fmaxf(y2, 0.0f), IMG_H - 1.0f);
      size_t o = (size_t)(b * NC1 + (c - 1)) * NPROP + n;
      scores_ws[o] = prob;
      ((float4*)boxes_ws)[o] = make_float4(x1, y1, x2, y2);
    }
  }
}

// ======================================================================
// Kernel B: one workgroup per (batch, class): sort -> valid-prefix ->
// greedy NMS over the V valid boxes (LDS-resident) -> masked outputs.
// ======================================================================
__global__ __launch_bounds__(TPB) void nms_kernel(
    const float* __restrict__ scores_ws,   // [160, 1000]
    const float* __restrict__ boxes_ws,    // [160, 1000, 4]
    float* __restrict__ out5,              // [160, 1000, 5]
    float* __restrict__ labels_out,        // [160, 1000]
    float* __restrict__ keep_out)          // [160, 1000]
{
  __shared__ __align__(16) float sraw[NPROP];      // raw scores (TDM dest)
  __shared__ float    skey[MSORT];
  __shared__ int      sidx[MSORT];
  __shared__ __align__(16) float sbox[NPROP * 4];  // boxes in sorted order
  __shared__ float    sarea[NPROP];
  __shared__ unsigned skeep[NPROP];                // keep flag per ORIGINAL idx
  __shared__ unsigned supp[MSORT / 32];            // suppression bitmask
  __shared__ int      sV;
  __shared__ float    cb[4];
  __shared__ int      ckeep;

  const int tid = threadIdx.x;
  const int cls = blockIdx.x;                       // = b*80 + (c-1)
  const float* gsc = scores_ws + (size_t)cls * NPROP;
  const float* gbx = boxes_ws  + (size_t)cls * NPROP * 4;

  // ---- Phase 0: TDM DMA of the 1000-score row into LDS (wave 0 issues;
  //      TDM ignores EXEC so the whole tile moves), others init LDS state.
  if (tid == 0) {
    unsigned long long ga = (unsigned long long)gsc;
    u32x4_t g0;                                   // D# group 0 (128b)
    g0[0] = 1u;                                   // count=1, no gather
    g0[1] = lds_off(sraw);                        // lds_addr
    g0[2] = (unsigned)ga;                         // global_addr[31:0]
    g0[3] = (unsigned)(ga >> 32) | (2u << 30);    // addr[56:32] | type=2
    u32x8_t g1;                                   // D# group 1 (256b)
    g1[0] = 2u << 16;                             // data_size=4B, mask=0
    g1[1] = (unsigned)NPROP << 16;                // tensor_dim0[15:0]=1000
    g1[2] = 1u << 16;                             // tensor_dim1=1
    g1[3] = (unsigned)NPROP << 16;                // tile_dim0=1000
    g1[4] = 1u;                                   // tile_dim1=1, tile_dim2=0
    g1[5] = (unsigned)NPROP;                      // tensor_dim0_stride=1000
    g1[6] = (unsigned)NPROP << 16;                // tensor_dim1_stride=1000
    g1[7] = 0u;
    asm volatile("tensor_load_to_lds %0, %1" :: "s"(g0), "s"(g1) : "memory");
  }
  for (int j = tid; j < NPROP; j += TPB) skeep[j] = 0u;
  if (tid < MSORT / 32) supp[tid] = 0u;
  if (tid == 0) sV = 0;
  for (int r = NPROP + tid; r < MSORT; r += TPB) { skey[r] = -3.0e38f; sidx[r] = r; }

  wait_tensor0();          // scalar wait; waves with TENSORcnt==0 pass through
  __syncthreads();

  // ---- Phase 1: sort keys (invalid => -1e30 so they sink; stable tiebreak)
  for (int j = tid; j < NPROP; j += TPB) {
    float sc = sraw[j];
    skey[j] = (sc > SCORE_THRESH) ? sc : -1.0e30f;
    sidx[j] = j;
  }

  // ---- Phase 2: bitonic sort, descending, 1024 elements
  for (unsigned k = 2; k <= MSORT; k <<= 1) {
    for (unsigned j = k >> 1; j > 0; j >>= 1) {
      __syncthreads();
      for (unsigned i = (unsigned)tid; i < MSORT; i += TPB) {
        unsigned ixj = i ^ j;
        if (ixj > i) {
          float ka = skey[i], kb = skey[ixj];
          int   ia = sidx[i], ib = sidx[ixj];
          bool beforeAB = (ka > kb) || (ka == kb && ia < ib);
          bool up = ((i & k) == 0);
          if (up ? !beforeAB : beforeAB) {
            skey[i] = kb; skey[ixj] = ka;
            sidx[i] = ib; sidx[ixj] = ia;
          }
        }
      }
    }
  }
  __syncthreads();

  // ---- Phase 3: V = length of valid (score > thresh) prefix
  for (int r = tid; r < NPROP; r += TPB) {
    bool v0 = skey[r] > SCORE_THRESH;
    bool v1 = (r + 1 < NPROP) && (skey[r + 1] > SCORE_THRESH);
    if (v0 && !v1) sV = r + 1;                    // unique boundary writer
  }
  __syncthreads();
  const int V = sV;                               // uniform

  // ---- Phase 4: async per-lane gather of the V sorted boxes into LDS
  for (int r = tid; r < V; r += TPB)
    async_gather_b128(gbx + 4 * sidx[r], &sbox[4 * r]);
  wait_async0();
  __syncthreads();
  for (int r = tid; r < V; r += TPB)
    sarea[r] = (sbox[4*r+2] - sbox[4*r+0] + 1.0f) *
               (sbox[4*r+3] - sbox[4*r+1] + 1.0f);
  __syncthreads();

  // ---- Phase 5: greedy NMS over the V valid boxes (V ~ 30 typical)
  for (int i = 0; i < V; ++i) {
    if (tid == 0) {
      int kept = !((supp[i >> 5] >> (i & 31)) & 1u);
      ckeep = kept;
      if (kept) {
        skeep[sidx[i]] = 1u;
        cb[0] = sbox[4*i+0]; cb[1] = sbox[4*i+1];
        cb[2] = sbox[4*i+2]; cb[3] = sbox[4*i+3];
      }
    }
    __syncthreads();
    if (ckeep) {
      float ix1 = cb[0], iy1 = cb[1], ix2 = cb[2], iy2 = cb[3];
      float ia = (ix2 - ix1 + 1.0f) * (iy2 - iy1 + 1.0f);
      for (int j = i + 1 + tid; j < V; j += TPB) {
        float xx1 = fmaxf(ix1, sbox[4*j+0]);
        float yy1 = fmaxf(iy1, sbox[4*j+1]);
        float xx2 = fminf(ix2, sbox[4*j+2]);
        float yy2 = fminf(iy2, sbox[4*j+3]);
        float ww = fmaxf(xx2 - xx1 + 1.0f, 0.0f);
        float hh = fmaxf(yy2 - yy1 + 1.0f, 0.0f);
        float inter = ww * hh;
        float iou = inter / (ia + sarea[j] - inter);
        if (iou > NMS_THRESH) atomicOr(&supp[j >> 5], 1u << (j & 31));  // ds_or
      }
    }
    __syncthreads();
  }
  __syncthreads();

  // ---- Phase 6: masked outputs (contiguous, coalesced)
  const float lbl = (float)((cls % NC1) + 1);
  for (int n = tid; n < NPROP; n += TPB) {
    float kf = skeep[n] ? 1.0f : 0.0f;
    float4 bx = ((const float4*)gbx)[n];
    float* op = out5 + ((size_t)cls * NPROP + n) * 5;
    op[0] = bx.x * kf; op[1] = bx.y * kf;
    op[2] = bx.z * kf; op[3] = bx.w * kf;
    op[4] = sraw[n] * kf;
    labels_out[(size_t)cls * NPROP + n] = lbl;
    keep_out[(size_t)cls * NPROP + n]   = kf;
  }
}

// ======================================================================
extern "C" void kernel_launch(void* const* d_in, const int* in_sizes, int n_in,
                              void* d_out, int out_size, void* d_ws, size_t ws_size,
                              hipStream_t stream) {
  (void)in_sizes; (void)n_in; (void)out_size; (void)ws_size;
  const float* logits = (const float*)d_in[0];   // [2000, 81]
  const float* regr   = (const float*)d_in[1];   // [2000, 324]
  const float* props  = (const float*)d_in[2];   // [2000, 4]

  float* scores_ws = (float*)d_ws;                                  // 160k f32
  float* boxes_ws  = scores_ws + (size_t)BATCH * NC1 * NPROP;       // 640k f32

  float* out5   = (float*)d_out;                                    // 800k
  float* labels = out5 + (size_t)BATCH * NC1 * NPROP * 5;           // 160k
  float* keep   = labels + (size_t)BATCH * NC1 * NPROP;             // 160k

  hipLaunchKernelGGL(decode_softmax_kernel, dim3((BATCH * NPROP + 3) / 4),
                     dim3(128), 0, stream, logits, regr, props,
                     scores_ws, boxes_ws);
  hipLaunchKernelGGL(nms_kernel, dim3(BATCH * NC1), dim3(TPB), 0, stream,
                     scores_ws, boxes_ws, out5, labels, keep);
}